// MyModel_89524298318328
// MI455X (gfx1250) — compile-verified
//
#include <hip/hip_runtime.h>
#include <math.h>

typedef __attribute__((ext_vector_type(2))) float v2f;
typedef __attribute__((ext_vector_type(8))) float v8f;

#define PI_F 3.14159265358979323846f

__device__ const float g_EXPS[8] = {5.447178f, 0.8245472f, 0.1831916f, 1.0f,
                                    5.447178f, 0.8245472f, 0.1831916f, 1.0f};
__device__ const float g_COEF[8] = {0.156285f, 0.904691f, 1.0f, 0.0f,
                                    0.156285f, 0.904691f, 1.0f, 0.0f};

// Boys function F0
__device__ __forceinline__ float f0_dev(float t) {
    float ts  = fmaxf(t, 1e-12f);
    float big = 0.5f * sqrtf(PI_F / ts) * erff(sqrtf(ts));
    return (t < 1e-10f) ? (1.0f - t * (1.0f / 3.0f)) : big;
}

// D = A(16x4) * B(4x16) + C  -- CDNA5 WMMA, f32, K=4
__device__ __forceinline__ v8f wmma4(v2f a, v2f b, v8f c) {
    return __builtin_amdgcn_wmma_f32_16x16x4_f32(
        /*neg_a=*/false, a, /*neg_b=*/false, b,
        /*c_mod=*/(short)0, c, /*reuse_a=*/false, /*reuse_b=*/false);
}

// Cyclic Jacobi eigensolver for symmetric 4x4; A -> diag(lam), V columns = eigenvectors.
// Fully unrolled inner loops: register arrays with static indices only.
__device__ void jacobi4(float* A, float* V, float* lam) {
    #pragma unroll
    for (int i = 0; i < 4; i++)
        #pragma unroll
        for (int j = 0; j < 4; j++) V[i*4+j] = (i == j) ? 1.0f : 0.0f;
    for (int sweep = 0; sweep < 14; ++sweep) {
        #pragma unroll
        for (int p = 0; p < 3; ++p) {
            #pragma unroll
            for (int q = p + 1; q < 4; ++q) {
                float apq = A[p*4+q];
                if (fabsf(apq) > 1e-20f) {
                    float app = A[p*4+p], aqq = A[q*4+q];
                    float tau = (aqq - app) / (2.0f * apq);
                    float t   = ((tau >= 0.0f) ? 1.0f : -1.0f) /
                                (fabsf(tau) + sqrtf(1.0f + tau * tau));
                    float cth = 1.0f / sqrtf(1.0f + t * t);
                    float sth = t * cth;
                    #pragma unroll
                    for (int k = 0; k < 4; ++k) {       // A = A*G
                        float akp = A[k*4+p], akq = A[k*4+q];
                        A[k*4+p] = cth * akp - sth * akq;
                        A[k*4+q] = sth * akp + cth * akq;
                    }
                    #pragma unroll
                    for (int k = 0; k < 4; ++k) {       // A = G^T*A
                        float apk = A[p*4+k], aqk = A[q*4+k];
                        A[p*4+k] = cth * apk - sth * aqk;
                        A[q*4+k] = sth * apk + cth * aqk;
                    }
                    #pragma unroll
                    for (int k = 0; k < 4; ++k) {       // V = V*G
                        float vkp = V[k*4+p], vkq = V[k*4+q];
                        V[k*4+p] = cth * vkp - sth * vkq;
                        V[k*4+q] = sth * vkp + cth * vkq;
                    }
                }
            }
        }
    }
    #pragma unroll
    for (int i = 0; i < 4; i++) lam[i] = A[i*4+i];
}

__global__ void __launch_bounds__(32) hf_scf_kernel(const float* __restrict__ Rin,
                                                    float* __restrict__ out) {
    __shared__ float aexpL[8], crawL[8], cnormL[8], cenL[12];
    __shared__ float Slds[16], Hclds[16];
    __shared__ float pairP[64], pairKC[64], pairPx[64], pairPy[64], pairPz[64];
    __shared__ float ERIlds[256], Glds[256];
    __shared__ float Xlds[16], Flds[16], N1lds[16], Fplds[16], Dlds[16];

    const int lane = threadIdx.x;
    const int lm   = lane & 15;              // N (or M) index within half-wave
    const int lk   = (lane < 16) ? 0 : 2;    // base K slot for A/B fragments

    // ---- Step A: primitive exponents / raw coefficients / centers ----
    if (lane < 8) {
        float aa = g_EXPS[lane];
        aexpL[lane] = aa;
        crawL[lane] = g_COEF[lane] * powf(2.0f * aa / PI_F, 0.75f);
    }
    if (lane < 12) {
        int b = lane / 3, d = lane - 3 * b;
        int atom = b >> 1;                   // basis 0,1 -> atom 0 ; 2,3 -> atom 1
        cenL[lane] = Rin[atom * 3 + d];
    }
    __syncthreads();
    // normalization: c = craw / sqrt(diag(Sc)) ; Rii=0 so K=1
    if (lane < 8) {
        int i = lane >> 1;
        float sc = 0.0f;
        for (int p = 0; p < 2; p++)
            for (int q = 0; q < 2; q++) {
                float pv = aexpL[i*2+p] + aexpL[i*2+q];
                sc += crawL[i*2+p] * crawL[i*2+q] * powf(PI_F / pv, 1.5f);
            }
        cnormL[lane] = crawL[lane] * rsqrtf(sc);
    }
    __syncthreads();

    // ---- Step B: 64 primitive-pair quantities (p, c*c*K, P vector) ----
    for (int id = lane; id < 64; id += 32) {
        int u = id >> 3, v = id & 7;
        int i = u >> 1, p = u & 1, j = v >> 1, q = v & 1;
        float ai = aexpL[i*2+p], aj = aexpL[j*2+q];
        float pv = ai + aj;
        float mu = ai * aj / pv;
        float dx = cenL[i*3+0] - cenL[j*3+0];
        float dy = cenL[i*3+1] - cenL[j*3+1];
        float dz = cenL[i*3+2] - cenL[j*3+2];
        float r2 = dx*dx + dy*dy + dz*dz;
        float Kf = expf(-mu * r2);
        pairP[id]  = pv;
        pairKC[id] = cnormL[i*2+p] * cnormL[j*2+q] * Kf;
        pairPx[id] = (ai * cenL[i*3+0] + aj * cenL[j*3+0]) / pv;
        pairPy[id] = (ai * cenL[i*3+1] + aj * cenL[j*3+1]) / pv;
        pairPz[id] = (ai * cenL[i*3+2] + aj * cenL[j*3+2]) / pv;
    }

    // ---- Step C: S and Hc = T + V (one (i,j) entry per lane) ----
    if (lane < 16) {
        int i = lane >> 2, j = lane & 3;
        float sS = 0.0f, sT = 0.0f, sV = 0.0f;
        for (int p = 0; p < 2; p++)
            for (int q = 0; q < 2; q++) {
                float ai = aexpL[i*2+p], aj = aexpL[j*2+q];
                float pv = ai + aj, mu = ai * aj / pv;
                float dx = cenL[i*3+0] - cenL[j*3+0];
                float dy = cenL[i*3+1] - cenL[j*3+1];
                float dz = cenL[i*3+2] - cenL[j*3+2];
                float r2 = dx*dx + dy*dy + dz*dz;
                float Kf = expf(-mu * r2);
                float sprim = powf(PI_F / pv, 1.5f) * Kf;
                float tprim = mu * (3.0f - 2.0f * mu * r2) * sprim;
                float Px = (ai * cenL[i*3+0] + aj * cenL[j*3+0]) / pv;
                float Py = (ai * cenL[i*3+1] + aj * cenL[j*3+1]) / pv;
                float Pz = (ai * cenL[i*3+2] + aj * cenL[j*3+2]) / pv;
                float vsum = 0.0f;
                for (int a2 = 0; a2 < 2; a2++) {      // both Z = 1
                    float cx = Px - Rin[a2*3+0], cy = Py - Rin[a2*3+1], cz = Pz - Rin[a2*3+2];
                    vsum += f0_dev(pv * (cx*cx + cy*cy + cz*cz));
                }
                float vprim = -(2.0f * PI_F / pv) * Kf * vsum;
                float cc = cnormL[i*2+p] * cnormL[j*2+q];
                sS += cc * sprim; sT += cc * tprim; sV += cc * vprim;
            }
        Slds[lane]  = sS;
        Hclds[lane] = sT + sV;
    }
    __syncthreads();

    // ---- Step D: ERI tensor (8 entries per lane, 16 primitive quartets each) ----
    for (int e = 0; e < 8; e++) {
        int idx = lane * 8 + e;
        int l = idx & 3, k = (idx >> 2) & 3, j = (idx >> 4) & 3, i = (idx >> 6) & 3;
        float acc = 0.0f;
        for (int p = 0; p < 2; p++) for (int q = 0; q < 2; q++)
        for (int r = 0; r < 2; r++) for (int s = 0; s < 2; s++) {
            int m2 = (i*2+p) * 8 + (j*2+q);
            int n2 = (k*2+r) * 8 + (l*2+s);
            float pm = pairP[m2], pn = pairP[n2];
            float sm = pm + pn;
            float dx = pairPx[m2] - pairPx[n2];
            float dy = pairPy[m2] - pairPy[n2];
            float dz = pairPz[m2] - pairPz[n2];
            float alpha = pm * pn / sm;
            // 2*pi^2.5 = 34.98683665...
            acc += pairKC[m2] * pairKC[n2] *
                   (34.98683665524972f / (pm * pn * sqrtf(sm))) *
                   f0_dev(alpha * (dx*dx + dy*dy + dz*dz));
        }
        ERIlds[idx] = acc;
    }
    __syncthreads();

    // ---- Step E: effective 2-body matrix G[(ij),(kl)] = ERI_ijkl - 0.5*ERI_ikjl ----
    for (int e = 0; e < 8; e++) {
        int g = lane * 8 + e;
        int col = g & 15, row = g >> 4;
        int i = row >> 2, j = row & 3, k = col >> 2, l = col & 3;
        Glds[g] = ERIlds[g] - 0.5f * ERIlds[((i*4+k)*4+j)*4+l];
    }
    __syncthreads();

    // ---- Step F: eigh(S) -> X = V diag(lam^-1/2) V^T (symmetric), all lanes redundant ----
    float Xr[16];
    {
        float As[16], Vs[16], lamS[4];
        #pragma unroll
        for (int t = 0; t < 16; t++) As[t] = Slds[t];
        jacobi4(As, Vs, lamS);
        float ir[4];
        #pragma unroll
        for (int t = 0; t < 4; t++) ir[t] = rsqrtf(lamS[t]);
        #pragma unroll
        for (int a = 0; a < 4; a++)
            #pragma unroll
            for (int b = 0; b < 4; b++) {
                float x = 0.0f;
                #pragma unroll
                for (int t = 0; t < 4; t++) x += Vs[a*4+t] * ir[t] * Vs[b*4+t];
                Xr[a*4+b] = x;
            }
    }
    if (lane == 0) {
        #pragma unroll
        for (int t = 0; t < 16; t++) Xlds[t] = Xr[t];
        #pragma unroll
        for (int t = 0; t < 16; t++) Dlds[t] = 0.0f;   // D0 = 0
    }
    __syncthreads();

    // ---- loop-invariant WMMA fragments ----
    // A-fragments of G (full 16 rows, K split into 4 chunks of 4)
    float ga0[4], ga1[4];
    #pragma unroll
    for (int kk = 0; kk < 4; kk++) {
        ga0[kk] = Glds[lm*16 + 4*kk + lk];
        ga1[kk] = Glds[lm*16 + 4*kk + lk + 1];
    }
    // X as A-frag (16x4 padded) and B-frag (4x16 padded)
    v2f xa, xb;
    xa.x = (lm < 4) ? Xlds[lm*4 + lk]       : 0.0f;
    xa.y = (lm < 4) ? Xlds[lm*4 + lk + 1]   : 0.0f;
    xb.x = (lm < 4) ? Xlds[lk*4 + lm]       : 0.0f;
    xb.y = (lm < 4) ? Xlds[(lk+1)*4 + lm]   : 0.0f;
    // Hc embedded in column 0 of a C-fragment
    v8f hcf = {0.f,0.f,0.f,0.f,0.f,0.f,0.f,0.f};
    if (lane == 0) {
        #pragma unroll
        for (int r = 0; r < 8; r++) hcf[r] = Hclds[r];
    }
    if (lane == 16) {
        #pragma unroll
        for (int r = 0; r < 8; r++) hcf[r] = Hclds[8 + r];
    }
    const v8f vzero = {0.f,0.f,0.f,0.f,0.f,0.f,0.f,0.f};

    // ---- SCF loop: 30 density updates + final Fock build ----
    for (int it = 0; it < 31; ++it) {
        __syncthreads();                     // Dlds visible
        // F = Hc + G @ vec(D): 4 chained K=4 WMMAs, result in column 0
        v8f acc = hcf;
        #pragma unroll
        for (int kk = 0; kk < 4; kk++) {
            v2f a; a.x = ga0[kk]; a.y = ga1[kk];
            v2f b;
            b.x = (lm == 0) ? Dlds[4*kk + lk]     : 0.0f;
            b.y = (lm == 0) ? Dlds[4*kk + lk + 1] : 0.0f;
            acc = wmma4(a, b, acc);
        }
        if (lane == 0) {
            #pragma unroll
            for (int r = 0; r < 8; r++) Flds[r] = acc[r];
        }
        if (lane == 16) {
            #pragma unroll
            for (int r = 0; r < 8; r++) Flds[8 + r] = acc[r];
        }
        __syncthreads();
        if (it == 30) break;                 // final F ready for energy

        // N1 = F @ X  (F staged as A-frag from LDS, X B-frag invariant)
        v2f fa;
        fa.x = (lm < 4) ? Flds[lm*4 + lk]     : 0.0f;
        fa.y = (lm < 4) ? Flds[lm*4 + lk + 1] : 0.0f;
        v8f n1 = wmma4(fa, xb, vzero);
        if (lane < 4) {
            #pragma unroll
            for (int r = 0; r < 4; r++) N1lds[r*4 + lane] = n1[r];
        }
        __syncthreads();

        // Fp = X @ N1  (X A-frag invariant, N1 staged as B-frag)
        v2f nb;
        nb.x = (lm < 4) ? N1lds[lk*4 + lm]     : 0.0f;
        nb.y = (lm < 4) ? N1lds[(lk+1)*4 + lm] : 0.0f;
        v8f fp = wmma4(xa, nb, vzero);
        if (lane < 4) {
            #pragma unroll
            for (int r = 0; r < 4; r++) Fplds[r*4 + lane] = fp[r];
        }
        __syncthreads();

        // eigh(Fp) (redundant per lane), lowest eigenvector -> D = 2 c0 c0^T
        float Af[16], Vf[16], lamF[4];
        #pragma unroll
        for (int t = 0; t < 16; t++) Af[t] = Fplds[t];
        jacobi4(Af, Vf, lamF);
        int mmin = 0; float lmin = lamF[0];
        #pragma unroll
        for (int t = 1; t < 4; t++) { if (lamF[t] < lmin) { lmin = lamF[t]; mmin = t; } }
        float cp[4];
        #pragma unroll
        for (int t = 0; t < 4; t++) {
            float v0 = Vf[t*4+0], v1 = Vf[t*4+1], v2 = Vf[t*4+2], v3 = Vf[t*4+3];
            cp[t] = (mmin == 0) ? v0 : ((mmin == 1) ? v1 : ((mmin == 2) ? v2 : v3));
        }
        float c0[4];
        #pragma unroll
        for (int a = 0; a < 4; a++) {
            float x = 0.0f;
            #pragma unroll
            for (int b = 0; b < 4; b++) x += Xr[a*4+b] * cp[b];
            c0[a] = x;
        }
        if (lane == 0) {
            #pragma unroll
            for (int a = 0; a < 4; a++)
                #pragma unroll
                for (int b = 0; b < 4; b++) Dlds[a*4+b] = 2.0f * c0[a] * c0[b];
        }
    }

    // ---- energy: e = 0.5*sum(D*(Hc+F)) + 1/|R0-R1| ----
    if (lane == 0) {
        float e = 0.0f;
        #pragma unroll
        for (int t = 0; t < 16; t++) e += Dlds[t] * (Hclds[t] + Flds[t]);
        e *= 0.5f;
        float dx = Rin[0] - Rin[3], dy = Rin[1] - Rin[4], dz = Rin[2] - Rin[5];
        e += 1.0f / sqrtf(dx*dx + dy*dy + dz*dz);
        out[0] = e;
    }
}

extern "C" void kernel_launch(void* const* d_in, const int* in_sizes, int n_in,
                              void* d_out, int out_size, void* d_ws, size_t ws_size,
                              hipStream_t stream) {
    (void)in_sizes; (void)n_in; (void)out_size; (void)d_ws; (void)ws_size;
    const float* R = (const float*)d_in[0];
    float* out = (float*)d_out;
    hipLaunchKernelGGL(hf_scf_kernel, dim3(1), dim3(32), 0, stream, R, out);
}